// SelfAttention_23210003268270
// MI455X (gfx1250) — compile-verified
//
#include <hip/hip_runtime.h>
#include <hip/hip_bf16.h>

typedef float v2f __attribute__((ext_vector_type(2)));
typedef float v8f __attribute__((ext_vector_type(8)));

static __device__ __forceinline__ v8f wmma_f32_16x16x4(v2f a, v2f b, v8f c) {
    // D = A(16x4) * B(4x16) + C, full fp32
    return __builtin_amdgcn_wmma_f32_16x16x4_f32(
        /*neg_a=*/false, a, /*neg_b=*/false, b,
        /*c_mod=*/(short)0, c, /*reuse_a=*/false, /*reuse_b=*/false);
}

// ---- CDNA5 async global->LDS (ASYNCcnt) path, guarded for toolchain drift ----
#if __has_builtin(__builtin_amdgcn_global_load_async_to_lds_b128)
#define HAS_ASYNC_LDS 1
typedef int i32x4 __attribute__((vector_size(16)));
typedef __attribute__((address_space(1))) i32x4 g_i32x4;   // global
typedef __attribute__((address_space(3))) i32x4 l_i32x4;   // LDS
#else
#define HAS_ASYNC_LDS 0
#endif

static __device__ __forceinline__ void wait_async_all() {
#if __has_builtin(__builtin_amdgcn_s_wait_asynccnt)
    __builtin_amdgcn_s_wait_asynccnt(0);
#else
    asm volatile("s_wait_asynccnt 0x0" ::: "memory");
#endif
}

// ---------------------------------------------------------------------------
// Kernel 1: QKV projections.  P = X @ W  for W in {Wq, Wk, Wv} (blockIdx.z).
// Block = 256 threads (8 waves). Block tile 128x64, wave tile 32x32 (2x2 frags).
// Double-buffered LDS tiles filled with GLOBAL_LOAD_ASYNC_TO_LDS_B128 so the
// next tile's global->LDS DMA overlaps the current tile's 32 WMMAs.
// ---------------------------------------------------------------------------
__global__ __launch_bounds__(256) void qkv_proj_kernel(
    const float* __restrict__ X,
    const float* __restrict__ Wq, const float* __restrict__ Wk,
    const float* __restrict__ Wv,
    float* __restrict__ Qo, float* __restrict__ Ko, float* __restrict__ Vo)
{
    constexpr int D   = 1024;
    constexpr int BM  = 128;
    constexpr int BN  = 64;
    constexpr int KT  = 32;
    constexpr int LDX = 36;   // 32 + 4 pad: 16B-aligned float4 slots, distinct banks
    constexpr int LDW = 80;   // 64 + 16 pad: lane-halves on disjoint bank groups

    const float* W;
    float* O;
    if (blockIdx.z == 0)      { W = Wq; O = Qo; }
    else if (blockIdx.z == 1) { W = Wk; O = Ko; }
    else                      { W = Wv; O = Vo; }

    __shared__ float xt[2][BM * LDX];   // 2 x 18 KB
    __shared__ float wt[2][KT * LDW];   // 2 x 10 KB

    const int tid  = threadIdx.x;
    const int lane = tid & 31;
    const int wid  = tid >> 5;
    const int l    = lane & 15;
    const int half = lane >> 4;
    const int wr   = wid >> 1;      // 0..3
    const int wc   = wid & 1;       // 0..1

    const int row0 = blockIdx.x * BM;
    const int col0 = blockIdx.y * BN;

    // stage one K-chunk (X tile 128x32, W tile 32x64) into LDS buffer `buf`
    auto stage = [&](int buf, int k0) {
        #pragma unroll
        for (int i = 0; i < 4; ++i) {
            int idx4 = tid + i * 256;
            int r  = idx4 >> 3;
            int c4 = (idx4 & 7) << 2;
            const float* gp = &X[(size_t)(row0 + r) * D + k0 + c4];
            float*       lp = &xt[buf][r * LDX + c4];
#if HAS_ASYNC_LDS
            __builtin_amdgcn_global_load_async_to_lds_b128(
                (g_i32x4*)gp, (l_i32x4*)lp, 0, 0);
#else
            *(float4*)lp = *(const float4*)gp;
#endif
        }
        #pragma unroll
        for (int i = 0; i < 2; ++i) {
            int idx4 = tid + i * 256;
            int r  = idx4 >> 4;
            int c4 = (idx4 & 15) << 2;
            const float* gp = &W[(size_t)(k0 + r) * D + col0 + c4];
            float*       lp = &wt[buf][r * LDW + c4];
#if HAS_ASYNC_LDS
            __builtin_amdgcn_global_load_async_to_lds_b128(
                (g_i32x4*)gp, (l_i32x4*)lp, 0, 0);
#else
            *(float4*)lp = *(const float4*)gp;
#endif
        }
    };

    v8f acc[2][2] = {};

    stage(0, 0);
    for (int kc = 0; kc < D / KT; ++kc) {
        const int cur = kc & 1;
#if HAS_ASYNC_LDS
        wait_async_all();            // this wave's DMA into buf[cur] done
#endif
        __syncthreads();             // all waves' fills done; prev reads of nxt done
        if (kc + 1 < D / KT)
            stage(cur ^ 1, (kc + 1) * KT);   // overlap next tile's DMA with WMMAs

        #pragma unroll
        for (int ks = 0; ks < KT; ks += 4) {
            v2f a[2], b[2];
            #pragma unroll
            for (int i = 0; i < 2; ++i)
                a[i] = *(const v2f*)&xt[cur][(wr * 32 + 16 * i + l) * LDX + ks + 2 * half];
            #pragma unroll
            for (int j = 0; j < 2; ++j) {
                v2f t;
                t.x = wt[cur][(ks + 2 * half + 0) * LDW + wc * 32 + 16 * j + l];
                t.y = wt[cur][(ks + 2 * half + 1) * LDW + wc * 32 + 16 * j + l];
                b[j] = t;
            }
            #pragma unroll
            for (int i = 0; i < 2; ++i)
                #pragma unroll
                for (int j = 0; j < 2; ++j)
                    acc[i][j] = wmma_f32_16x16x4(a[i], b[j], acc[i][j]);
        }
        __syncthreads();             // reads of buf[cur] done before it refills
    }

    // C/D layout: vgpr r -> row (r + 8*half), col = l
    #pragma unroll
    for (int i = 0; i < 2; ++i)
        #pragma unroll
        for (int j = 0; j < 2; ++j)
            #pragma unroll
            for (int r = 0; r < 8; ++r) {
                int grow = row0 + wr * 32 + 16 * i + r + 8 * half;
                int gcol = col0 + wc * 32 + 16 * j + l;
                O[(size_t)grow * D + gcol] = acc[i][j][r];
            }
}

// ---------------------------------------------------------------------------
// Kernel 2: causal flash attention, fp32 throughout.
// One block (256 thr / 8 waves) per 16 query rows. Wave `wid` owns
//   - K-slab [wid*128, wid*128+128) for the QK^T partial reduction
//   - V-column slab [wid*128, ...)  -> 8 f32 16x16 accum fragments.
// Online softmax (m, l, alpha) in LDS; P tile re-read as A fragments for P@V.
// Next key-tile K/V rows prefetched (global_prefetch_b8) during current WMMAs.
// ---------------------------------------------------------------------------
__global__ __launch_bounds__(256) void flash_attn_kernel(
    const float* __restrict__ Q, const float* __restrict__ K,
    const float* __restrict__ V, float* __restrict__ O)
{
    constexpr int D     = 1024;
    constexpr int LDS_S = 18;            // 16 + 2 pad -> halves on disjoint banks

    __shared__ float spart[8 * 16 * LDS_S];  // per-wave QK^T partials
    __shared__ float sbuf[16 * LDS_S];       // reduced logits, then P tile
    __shared__ float mrow[16], lrow[16], arow[16];

    const int tid  = threadIdx.x;
    const int lane = tid & 31;
    const int wid  = tid >> 5;
    const int l    = lane & 15;
    const int half = lane >> 4;

    const int qt = blockIdx.x;      // query tile index, 0..255
    const int q0 = qt * 16;
    const int kslab   = wid * 128;  // this wave's K-reduction slab
    const int colbase = wid * 128;  // this wave's V-column slab

    if (tid < 16) { mrow[tid] = -__builtin_inff(); lrow[tid] = 0.0f; }
    __syncthreads();

    v8f acc[8] = {};

    for (int j = 0; j <= qt; ++j) {
        const int key0 = j * 16;

        // warm L0/L2 for the next key tile while this one computes
        if (j < qt) {
            const int nk = key0 + 16;
            __builtin_prefetch(&K[(size_t)(nk + l) * D + kslab   + half * 64], 0, 3);
            __builtin_prefetch(&V[(size_t)(nk + l) * D + colbase + half * 64], 0, 3);
        }

        // ---- S partial = Q[q0:+16, kslab:+128] @ K[key0:+16, kslab:+128]^T
        v8f sp = {};
        #pragma unroll 8
        for (int kk = 0; kk < 128; kk += 4) {
            v2f a = *(const v2f*)&Q[(size_t)(q0 + l) * D + kslab + kk + 2 * half];
            v2f b = *(const v2f*)&K[(size_t)(key0 + l) * D + kslab + kk + 2 * half];
            sp = wmma_f32_16x16x4(a, b, sp);
        }
        #pragma unroll
        for (int r = 0; r < 8; ++r)
            spart[wid * (16 * LDS_S) + (r + 8 * half) * LDS_S + l] = sp[r];
        __syncthreads();

        // ---- reduce 8 partials, scale 1/sqrt(1024), apply causal mask
        {
            int row = tid >> 4, col = tid & 15;
            float s = 0.0f;
            #pragma unroll
            for (int p = 0; p < 8; ++p)
                s += spart[p * (16 * LDS_S) + row * LDS_S + col];
            s *= 0.03125f;
            if (key0 + col > q0 + row) s = -__builtin_inff();
            sbuf[row * LDS_S + col] = s;
        }
        __syncthreads();

        // ---- online softmax bookkeeping (16 row threads)
        if (tid < 16) {
            int row = tid;
            float m = mrow[row], mn = m;
            #pragma unroll
            for (int c = 0; c < 16; ++c) mn = fmaxf(mn, sbuf[row * LDS_S + c]);
            float alpha = __expf(m - mn);
            float ls = 0.0f;
            #pragma unroll
            for (int c = 0; c < 16; ++c) {
                float p = __expf(sbuf[row * LDS_S + c] - mn);
                sbuf[row * LDS_S + c] = p;     // S -> P in place
                ls += p;
            }
            mrow[row] = mn;
            arow[row] = alpha;
            lrow[row] = lrow[row] * alpha + ls;
        }
        __syncthreads();

        // ---- rescale accumulators by alpha(row)
        float av[8];
        #pragma unroll
        for (int r = 0; r < 8; ++r) av[r] = arow[r + 8 * half];
        #pragma unroll
        for (int f = 0; f < 8; ++f)
            #pragma unroll
            for (int r = 0; r < 8; ++r) acc[f][r] *= av[r];

        // ---- acc += P(16x16) @ V[key0:+16, colbase:+128]
        #pragma unroll
        for (int t = 0; t < 4; ++t) {
            v2f a = *(const v2f*)&sbuf[l * LDS_S + 4 * t + 2 * half];
            #pragma unroll
            for (int f = 0; f < 8; ++f) {
                const float* vp =
                    &V[(size_t)(key0 + 4 * t + 2 * half) * D + colbase + 16 * f + l];
                v2f b;
                b.x = vp[0];
                b.y = vp[D];
                acc[f] = wmma_f32_16x16x4(a, b, acc[f]);
            }
        }
        __syncthreads();   // protect sbuf/arow before next iteration rewrites
    }

    // ---- normalize by running row sum and store
    float inv[8];
    #pragma unroll
    for (int r = 0; r < 8; ++r) inv[r] = 1.0f / lrow[r + 8 * half];
    #pragma unroll
    for (int f = 0; f < 8; ++f)
        #pragma unroll
        for (int r = 0; r < 8; ++r) {
            int grow = q0 + r + 8 * half;
            int gcol = colbase + 16 * f + l;
            O[(size_t)grow * D + gcol] = acc[f][r] * inv[r];
        }
}

// ---------------------------------------------------------------------------
extern "C" void kernel_launch(void* const* d_in, const int* in_sizes, int n_in,
                              void* d_out, int out_size, void* d_ws, size_t ws_size,
                              hipStream_t stream)
{
    const float* x  = (const float*)d_in[0];
    const float* Wq = (const float*)d_in[1];
    const float* Wk = (const float*)d_in[2];
    const float* Wv = (const float*)d_in[3];
    float* out = (float*)d_out;

    const size_t mat = (size_t)4096 * 1024;
    if (ws_size < 3 * mat * sizeof(float)) return;  // need 48 MB scratch

    float* Qw = (float*)d_ws;
    float* Kw = Qw + mat;
    float* Vw = Kw + mat;

    dim3 blk(256);
    dim3 g_proj(4096 / 128, 1024 / 64, 3);
    qkv_proj_kernel<<<g_proj, blk, 0, stream>>>(x, Wq, Wk, Wv, Qw, Kw, Vw);

    dim3 g_attn(4096 / 16);
    flash_attn_kernel<<<g_attn, blk, 0, stream>>>(Qw, Kw, Vw, out);
}